// PagedAttentionModel_10823317586421
// MI455X (gfx1250) — compile-verified
//
#include <hip/hip_runtime.h>
#include <hip/hip_bf16.h>
#include <math.h>

// ---------------- model constants (match reference) ----------------
#define BATCH   8
#define HIDDEN  4096
#define NH      32
#define NKV     8
#define GQA     4          // NH / NKV
#define HD      128
#define BLK     16
#define MAXB    256
#define TMAX    4096       // MAXB * BLK
#define FF      11008
#define RMS_EPS 1e-5f

typedef __attribute__((ext_vector_type(2))) float v2f;
typedef __attribute__((ext_vector_type(8))) float v8f;

// =====================================================================
// RMSNorm of rows 0..B-1 into a 16-row padded activation; pad rows = 0.
// grid = 16 blocks (one per padded row), 256 threads.
// =====================================================================
__global__ void rmsnorm_pad_kernel(const float* __restrict__ x,
                                   const float* __restrict__ w,
                                   float* __restrict__ out) {
  const int r = blockIdx.x;
  const int tid = threadIdx.x;
  __shared__ float red[256];
  if (r >= BATCH) {
    for (int i = tid; i < HIDDEN; i += 256) out[(size_t)r * HIDDEN + i] = 0.0f;
    return;
  }
  const float* xr = x + (size_t)r * HIDDEN;
  float s = 0.0f;
  for (int i = tid; i < HIDDEN; i += 256) { float v = xr[i]; s += v * v; }
  red[tid] = s; __syncthreads();
  for (int st = 128; st > 0; st >>= 1) {
    if (tid < st) red[tid] += red[tid + st];
    __syncthreads();
  }
  const float inv = rsqrtf(red[0] * (1.0f / HIDDEN) + RMS_EPS);
  for (int i = tid; i < HIDDEN; i += 256)
    out[(size_t)r * HIDDEN + i] = xr[i] * inv * w[i];
}

// =====================================================================
// WMMA fp32 GEMM: D[16,N] = A[16,K] * W[K,N], K-split into S slices.
// One wave (32 threads) per (n-tile, k-slice). Partials: [S][8][N].
// K loop unrolled 4x: all 8 fragments loaded in one clause before the
// dependent 4-WMMA accumulate chain (4x fewer waits, 4x more MLP).
// A fragment (V_WMMA_F32_16X16X4_F32, ISA 7.12.2): lanes 0-15 -> M rows,
// VGPR0/1 hold K={0,1} (lanes 0-15) / K={2,3} (lanes 16-31).
// B fragment mirrored: row K striped across lanes within a VGPR.
// Requires (K/S) % 16 == 0 (true for all GEMMs here: 512, 2752).
// =====================================================================
__global__ void gemm_wmma_kernel(const float* __restrict__ A,   // [16 x K]
                                 const float* __restrict__ W,   // [K x N]
                                 float* __restrict__ partial,   // [S x 8 x N]
                                 int K, int N, int S) {
  const int tile = blockIdx.x;       // column tile (16 cols)
  const int s    = blockIdx.y;       // K slice
  const int lane = threadIdx.x;      // wave32
  const int n0   = tile * 16;
  const int kPer = K / S;
  const int kBeg = s * kPer;
  const int kEnd = kBeg + kPer;
  const int half = lane >> 4;        // 0: K pair {0,1}, 1: K pair {2,3}
  const int l    = lane & 15;

  const float* Arow = A + (size_t)l * K;
  const float* Wp   = W + n0 + l;
  v8f c = {};

  for (int k0 = kBeg; k0 < kEnd; k0 += 16) {
    // stream-ahead prefetch of the weight panel (-> global_prefetch_b8)
    __builtin_prefetch(Wp + (size_t)(k0 + 64) * N, 0, 1);

    v2f a[4], b[4];
#pragma unroll
    for (int u = 0; u < 4; ++u) {
      const int ka = k0 + 4 * u + 2 * half;
      a[u] = *(const v2f*)(Arow + ka);            // one b64 load
      b[u].x = Wp[(size_t)ka * N];
      b[u].y = Wp[(size_t)(ka + 1) * N];
    }
#pragma unroll
    for (int u = 0; u < 4; ++u) {
      c = __builtin_amdgcn_wmma_f32_16x16x4_f32(
          /*neg_a=*/false, a[u], /*neg_b=*/false, b[u],
          /*c_mod=*/(short)0, c, /*reuse_a=*/false, /*reuse_b=*/false);
    }
  }

  // C/D layout: lanes 0-15 hold M=0..7 in VGPR 0..7 (exactly our B rows).
  if (lane < 16) {
    float* p = partial + (size_t)s * 8 * N + n0 + lane;
#pragma unroll
    for (int j = 0; j < 8; ++j) p[(size_t)j * N] = c[j];
  }
}

// out[i] = (residual ? residual[i] : 0) + sum_s partial[s][i], i < 8*N.
// Deterministic (fixed summation order) -> graph-replay bit-stable.
__global__ void reduce_partial_kernel(const float* __restrict__ partial,
                                      const float* __restrict__ residual,
                                      float* __restrict__ out, int N, int S) {
  const int i = blockIdx.x * 256 + threadIdx.x;
  const int total = 8 * N;
  if (i >= total) return;
  float acc = residual ? residual[i] : 0.0f;
  for (int s = 0; s < S; ++s) acc += partial[(size_t)s * total + i];
  out[i] = acc;
}

__global__ void fill_zero_kernel(float* __restrict__ p, int n) {
  const int i = blockIdx.x * 256 + threadIdx.x;
  if (i < n) p[i] = 0.0f;
}

// =====================================================================
// Paged-attention decode, one block per (kv-head, batch), 256 threads.
// Does NOT mutate the cache: token t == seqlen reads the fresh k/v.
// Phase A: wave-per-token q.k (float4 coalesced K rows, shfl reduce)
// Phase B: block softmax per group over t in [0, seqlen]
// Phase C: d-parallel attn.V (coalesced V rows)
// =====================================================================
__global__ void attn_decode_kernel(const float* __restrict__ q,      // [B, NH*HD]
                                   const float* __restrict__ knew,   // [B, NKV*HD]
                                   const float* __restrict__ vnew,   // [B, NKV*HD]
                                   const float* __restrict__ k_cache,
                                   const float* __restrict__ v_cache,
                                   const int*   __restrict__ block_table,
                                   const int*   __restrict__ cache_seqlens,
                                   float* __restrict__ scores,       // [B,NKV,GQA,TMAX]
                                   float* __restrict__ o_pad) {      // [16, NH*HD]
  const int kv  = blockIdx.x;
  const int b   = blockIdx.y;
  const int tid = threadIdx.x;
  const int seqlen = cache_seqlens[b];          // attend t in [0, seqlen]

  __shared__ float qs[GQA * HD];
  __shared__ float red[256];

  for (int i = tid; i < GQA * HD; i += 256)
    qs[i] = q[(size_t)b * (NH * HD) + (kv * GQA) * HD + i];
  __syncthreads();

  float* sc = scores + ((size_t)(b * NKV + kv)) * GQA * TMAX;
  const int* bt = block_table + b * MAXB;
  const int wave = tid >> 5, lane = tid & 31;
  const float scale = 0.08838834764831845f;     // 1/sqrt(128)

  // -------- Phase A: scores --------
  for (int t = wave; t <= seqlen; t += 8) {
    float4 kk;
    if (t == seqlen) {
      kk = *(const float4*)(knew + (size_t)b * (NKV * HD) + kv * HD + lane * 4);
    } else {
      const int blk = bt[t >> 4];
      const int off = t & (BLK - 1);
      kk = *(const float4*)(k_cache +
            (((size_t)blk * BLK + off) * NKV + kv) * HD + lane * 4);
    }
#pragma unroll
    for (int g = 0; g < GQA; ++g) {
      const float* qg = qs + g * HD + lane * 4;
      float p = qg[0] * kk.x + qg[1] * kk.y + qg[2] * kk.z + qg[3] * kk.w;
#pragma unroll
      for (int o = 16; o > 0; o >>= 1) p += __shfl_xor(p, o, 32);
      if (lane == 0) sc[g * TMAX + t] = p * scale;
    }
  }
  __syncthreads();

  // -------- Phase B: softmax per group --------
  for (int g = 0; g < GQA; ++g) {
    float m = -3.0e38f;
    for (int t = tid; t <= seqlen; t += 256) m = fmaxf(m, sc[g * TMAX + t]);
    red[tid] = m; __syncthreads();
    for (int st = 128; st > 0; st >>= 1) {
      if (tid < st) red[tid] = fmaxf(red[tid], red[tid + st]);
      __syncthreads();
    }
    m = red[0]; __syncthreads();

    float ssum = 0.0f;
    for (int t = tid; t <= seqlen; t += 256) {
      const float e = expf(sc[g * TMAX + t] - m);
      sc[g * TMAX + t] = e;
      ssum += e;
    }
    red[tid] = ssum; __syncthreads();
    for (int st = 128; st > 0; st >>= 1) {
      if (tid < st) red[tid] += red[tid + st];
      __syncthreads();
    }
    const float inv = 1.0f / red[0]; __syncthreads();
    for (int t = tid; t <= seqlen; t += 256) sc[g * TMAX + t] *= inv;
    __syncthreads();
  }

  // -------- Phase C: attn @ V --------
  if (tid < HD) {
    const int d = tid;
    float a0 = 0.f, a1 = 0.f, a2 = 0.f, a3 = 0.f;
    for (int t = 0; t <= seqlen; ++t) {
      float v;
      if (t == seqlen) {
        v = vnew[(size_t)b * (NKV * HD) + kv * HD + d];
      } else {
        const int blk = bt[t >> 4];
        const int off = t & (BLK - 1);
        v = v_cache[(((size_t)blk * BLK + off) * NKV + kv) * HD + d];
      }
      a0 += sc[0 * TMAX + t] * v;
      a1 += sc[1 * TMAX + t] * v;
      a2 += sc[2 * TMAX + t] * v;
      a3 += sc[3 * TMAX + t] * v;
    }
    float* op = o_pad + (size_t)b * (NH * HD) + (kv * GQA) * HD + d;
    op[0 * HD] = a0; op[1 * HD] = a1; op[2 * HD] = a2; op[3 * HD] = a3;
  }
}

// h = silu(g) * u over 8*FF elements (rows 0..7 of padded H)
__global__ void swiglu_kernel(const float* __restrict__ g,
                              const float* __restrict__ u,
                              float* __restrict__ h, int n) {
  const int i = blockIdx.x * 256 + threadIdx.x;
  if (i >= n) return;
  const float gv = g[i];
  h[i] = (gv / (1.0f + expf(-gv))) * u[i];
}

// =====================================================================
// Host-side launch
// =====================================================================
extern "C" void kernel_launch(void* const* d_in, const int* in_sizes, int n_in,
                              void* d_out, int out_size, void* d_ws, size_t ws_size,
                              hipStream_t stream) {
  (void)in_sizes; (void)n_in; (void)out_size; (void)ws_size;

  const float* x         = (const float*)d_in[0];
  const float* k_cache   = (const float*)d_in[1];
  const float* v_cache   = (const float*)d_in[2];
  const float* norm1_w   = (const float*)d_in[3];
  const float* wq        = (const float*)d_in[4];
  const float* wk        = (const float*)d_in[5];
  const float* wv        = (const float*)d_in[6];
  const float* wo        = (const float*)d_in[7];
  const float* norm2_w   = (const float*)d_in[8];
  const float* w_gate    = (const float*)d_in[9];
  const float* w_up      = (const float*)d_in[10];
  const float* w_down    = (const float*)d_in[11];
  const int*   block_tab = (const int*)d_in[12];
  const int*   seqlens   = (const int*)d_in[13];
  float* out = (float*)d_out;

  // workspace layout (floats)
  float* w = (float*)d_ws;
  float* A1   = w;                          // 16*4096
  float* Q    = A1  + 16 * HIDDEN;          // 8*4096
  float* Kn   = Q   + BATCH * NH * HD;      // 8*1024
  float* Vn   = Kn  + BATCH * NKV * HD;     // 8*1024
  float* O    = Vn  + BATCH * NKV * HD;     // 16*4096 (padded attn out)
  float* X1   = O   + 16 * HIDDEN;          // 8*4096
  float* A2   = X1  + BATCH * HIDDEN;       // 16*4096
  float* G    = A2  + 16 * HIDDEN;          // 8*11008
  float* U    = G   + BATCH * FF;           // 8*11008
  float* H    = U   + BATCH * FF;           // 16*11008 (padded)
  float* SC   = H   + 16 * FF;              // 8*8*4*4096 scores
  float* PART = SC  + (size_t)BATCH * NKV * GQA * TMAX; // max 8*8*11008

  const dim3 wave(32);
  const dim3 thr256(256);
  const int S1 = 8;   // K-split for K=4096 GEMMs (kPer=512, %16==0)
  const int S2 = 4;   // K-split for K=11008 GEMM (kPer=2752, %16==0)

  // 1) normed = rmsnorm(x, norm1_w), padded to 16 rows
  rmsnorm_pad_kernel<<<16, thr256, 0, stream>>>(x, norm1_w, A1);

  // 2) q / k / v projections (WMMA, K-split, deterministic reduce)
  gemm_wmma_kernel<<<dim3(NH * HD / 16, S1), wave, 0, stream>>>(A1, wq, PART, HIDDEN, NH * HD, S1);
  reduce_partial_kernel<<<(8 * NH * HD + 255) / 256, thr256, 0, stream>>>(PART, nullptr, Q, NH * HD, S1);

  gemm_wmma_kernel<<<dim3(NKV * HD / 16, S1), wave, 0, stream>>>(A1, wk, PART, HIDDEN, NKV * HD, S1);
  reduce_partial_kernel<<<(8 * NKV * HD + 255) / 256, thr256, 0, stream>>>(PART, nullptr, Kn, NKV * HD, S1);

  gemm_wmma_kernel<<<dim3(NKV * HD / 16, S1), wave, 0, stream>>>(A1, wv, PART, HIDDEN, NKV * HD, S1);
  reduce_partial_kernel<<<(8 * NKV * HD + 255) / 256, thr256, 0, stream>>>(PART, nullptr, Vn, NKV * HD, S1);

  // 3) paged attention decode (cache not mutated; new kv injected at t==seqlen)
  attn_decode_kernel<<<dim3(NKV, BATCH), thr256, 0, stream>>>(
      Q, Kn, Vn, k_cache, v_cache, block_tab, seqlens, SC, O);
  fill_zero_kernel<<<(8 * HIDDEN + 255) / 256, thr256, 0, stream>>>(O + (size_t)8 * HIDDEN, 8 * HIDDEN);

  // 4) x1 = x + o @ wo
  gemm_wmma_kernel<<<dim3(HIDDEN / 16, S1), wave, 0, stream>>>(O, wo, PART, NH * HD, HIDDEN, S1);
  reduce_partial_kernel<<<(8 * HIDDEN + 255) / 256, thr256, 0, stream>>>(PART, x, X1, HIDDEN, S1);

  // 5) h = rmsnorm(x1, norm2_w)
  rmsnorm_pad_kernel<<<16, thr256, 0, stream>>>(X1, norm2_w, A2);

  // 6) gate / up projections
  gemm_wmma_kernel<<<dim3(FF / 16, S1), wave, 0, stream>>>(A2, w_gate, PART, HIDDEN, FF, S1);
  reduce_partial_kernel<<<(8 * FF + 255) / 256, thr256, 0, stream>>>(PART, nullptr, G, FF, S1);

  gemm_wmma_kernel<<<dim3(FF / 16, S1), wave, 0, stream>>>(A2, w_up, PART, HIDDEN, FF, S1);
  reduce_partial_kernel<<<(8 * FF + 255) / 256, thr256, 0, stream>>>(PART, nullptr, U, FF, S1);

  // 7) H = silu(G) * U  (rows 0..7), pad rows zeroed
  swiglu_kernel<<<(8 * FF + 255) / 256, thr256, 0, stream>>>(G, U, H, 8 * FF);
  fill_zero_kernel<<<(8 * FF + 255) / 256, thr256, 0, stream>>>(H + (size_t)8 * FF, 8 * FF);

  // 8) out = x1 + H @ w_down
  gemm_wmma_kernel<<<dim3(HIDDEN / 16, S2), wave, 0, stream>>>(H, w_down, PART, FF, HIDDEN, S2);
  reduce_partial_kernel<<<(8 * HIDDEN + 255) / 256, thr256, 0, stream>>>(PART, X1, out, HIDDEN, S2);
}